// RotaryPositionalEmbedding_670014898751
// MI455X (gfx1250) — compile-verified
//
#include <hip/hip_runtime.h>
#include <math.h>

// RoPE on x:(8,16,4096,128) fp32. Memory-bound streaming kernel:
//   - wave32-coalesced 128-bit non-temporal loads/stores (one wave = one 512B row)
//   - cos/sin reuse: each thread fixes (position, quad) and loops over 32 of the
//     128 (batch*head) planes -> 2 sincos per 128 elements processed
//   - inv_freq table built once per block in double precision into LDS

typedef __attribute__((ext_vector_type(4))) float f32x4;

namespace {
constexpr int S       = 4096;       // sequence length
constexpr int D       = 128;        // head dim
constexpr int QUADS   = D / 4;      // 32 float4 per row
constexpr int PLANES  = 8 * 16;     // batch*heads
constexpr int PPT     = 32;         // planes processed per thread
constexpr int CHUNKS  = PLANES / PPT;               // 4
constexpr int NTHREADS = QUADS * S * CHUNKS;        // 524288
constexpr int BLOCK   = 256;                        // 8 waves (wave32)
}

__global__ __launch_bounds__(BLOCK) void rope_kernel(
    const float* __restrict__ x,
    const int*   __restrict__ tpos,   // token positions (arange, int32 on device)
    float*       __restrict__ out)
{
    // ---- per-block inv_freq table: 10000^(-p/64), p = pair index 0..63 ----
    __shared__ float freq[D / 2];
    const int lt = threadIdx.x;
    if (lt < D / 2) {
        // log2(10000) = 13.287712379549449; compute in f64, round to f32 so we
        // track the reference's fp32 pow as closely as possible.
        freq[lt] = (float)exp2(-(double)lt * (13.287712379549449 / 64.0));
    }
    __syncthreads();

    const unsigned tid   = blockIdx.x * (unsigned)BLOCK + threadIdx.x;
    const unsigned qid   = tid & (QUADS - 1);        // quad within row (0..31)
    const unsigned sid   = (tid >> 5) & (S - 1);     // sequence position
    const unsigned chunk = tid >> 17;                // which group of 32 planes

    // Angle per pair; this thread's float4 covers pairs p0=2*qid, p1=2*qid+1.
    const float pos = (float)tpos[sid];
    const float f0  = freq[2 * qid];
    const float f1  = freq[2 * qid + 1];
    float s0, c0, s1, c1;
    sincosf(pos * f0, &s0, &c0);
    sincosf(pos * f1, &s1, &c1);

    const size_t planeStride = (size_t)S * D;        // 524288 floats
    const size_t base = (size_t)(chunk * PPT) * planeStride
                      + (size_t)sid * D + (size_t)qid * 4;
    const float* xp = x   + base;
    float*       op = out + base;

    // Stream 32 planes: data touched exactly once -> non-temporal on both sides
    // so the 512MB stream doesn't churn L2.
#pragma unroll 8
    for (int p = 0; p < PPT; ++p) {
        const size_t off = (size_t)p * planeStride;
        f32x4 v = __builtin_nontemporal_load((const f32x4*)(xp + off));
        f32x4 r;
        r.x = c0 * v.x - s0 * v.y;   // even = cos*x1 - sin*x2
        r.y = s0 * v.x + c0 * v.y;   // odd  = sin*x1 + cos*x2
        r.z = c1 * v.z - s1 * v.w;
        r.w = s1 * v.z + c1 * v.w;
        __builtin_nontemporal_store(r, (f32x4*)(op + off));
    }
}

extern "C" void kernel_launch(void* const* d_in, const int* in_sizes, int n_in,
                              void* d_out, int out_size, void* d_ws, size_t ws_size,
                              hipStream_t stream) {
    (void)in_sizes; (void)n_in; (void)d_ws; (void)ws_size; (void)out_size;
    const float* x    = (const float*)d_in[0];
    const int*   tpos = (const int*)d_in[1];   // jnp int64 arange -> int32 under default JAX x64=off
    float*       out  = (float*)d_out;

    const int blocks = NTHREADS / BLOCK;       // 2048 blocks of 256 threads
    rope_kernel<<<blocks, BLOCK, 0, stream>>>(x, tpos, out);
}